// DQN_28080496181301
// MI455X (gfx1250) — compile-verified
//
#include <hip/hip_runtime.h>

typedef float v2f __attribute__((ext_vector_type(2)));
typedef float v8f __attribute__((ext_vector_type(8)));

#define HSZ   64
#define NOBS  256
#define DOBS  128
#define LSEQ  16
#define ADIM  40
#define F3IN  (NOBS*(DOBS+HSZ+NOBS))   // 114688
#define F3H   512
#define KSPLIT 56                       // 32 row-blocks x 56 = 1792 workgroups

// workspace layout in floats
#define WS_G   0
#define WS_E   64
#define WS_F   320
#define WS_X   1024
#define WS_H1  (WS_X + F3IN)            // 512-float raw accumulator

// ---------------------------------------------------------------------------
// Kernel 1: embedding gather + sequential GRU + rank-1 softmax factors.
// One workgroup (256 threads) — the whole stage is a few hundred K FLOPs.
// ---------------------------------------------------------------------------
__global__ __launch_bounds__(256) void k_encoder(
    const float* __restrict__ obs,  const int* __restrict__ tok,
    const float* __restrict__ embT, const float* __restrict__ Wih,
    const float* __restrict__ Whh,  const float* __restrict__ bih,
    const float* __restrict__ bhh,  const float* __restrict__ f1W,
    float* __restrict__ ws)
{
    __shared__ float s_emb[LSEQ * HSZ];
    __shared__ float s_h[HSZ];
    __shared__ float s_gi[3 * HSZ];
    __shared__ float s_gh[3 * HSZ];
    __shared__ float s_ab[2 * DOBS];
    __shared__ float s_u[NOBS];
    __shared__ float s_v[NOBS];
    __shared__ float s_red[256];

    const int t = threadIdx.x;

    for (int i = t; i < LSEQ * HSZ; i += 256)
        s_emb[i] = embT[tok[i / HSZ] * HSZ + (i % HSZ)];
    if (t < HSZ) s_h[t] = 0.0f;
    __syncthreads();

    // GRU, PyTorch gate order r,z,n
    for (int st = 0; st < LSEQ; ++st) {
        if (t < 3 * HSZ) {
            const float* xv = &s_emb[st * HSZ];
            float gi = bih[t], gh = bhh[t];
            for (int k = 0; k < HSZ; ++k) {
                gi += Wih[t * HSZ + k] * xv[k];
                gh += Whh[t * HSZ + k] * s_h[k];
            }
            s_gi[t] = gi; s_gh[t] = gh;
        }
        __syncthreads();
        if (t < HSZ) {
            float r = 1.0f / (1.0f + __expf(-(s_gi[t] + s_gh[t])));
            float z = 1.0f / (1.0f + __expf(-(s_gi[HSZ + t] + s_gh[HSZ + t])));
            float n = tanhf(s_gi[2 * HSZ + t] + r * s_gh[2 * HSZ + t]);
            s_h[t] = (1.0f - z) * n + z * s_h[t];
        }
        __syncthreads();
    }

    // a = Wa^T g (cols 0..127), b = Wb^T g (cols 128..255); f1W is [64, 256]
    if (t < 2 * DOBS) {
        float acc = 0.0f;
        for (int h = 0; h < HSZ; ++h) acc += f1W[h * (2 * DOBS) + t] * s_h[h];
        s_ab[t] = acc;
    }
    if (t < HSZ) ws[WS_G + t] = s_h[t];
    __syncthreads();

    // u_i = obs_i . a ; v_i = obs_i . b   (one row per thread)
    {
        float uu = 0.0f, vv = 0.0f;
        for (int d = 0; d < DOBS; ++d) {
            float o = obs[t * DOBS + d];
            uu += o * s_ab[d];
            vv += o * s_ab[DOBS + d];
        }
        s_u[t] = uu; s_v[t] = vv;
    }
    __syncthreads();

    // softmax over Z = u_i + v_j + c factorizes: p_ij = e_i * f_j
    // with e_i = exp(u_i-mu)/Su, f_j = exp(v_j-mv)/Sv  (constant c cancels)
    s_red[t] = s_u[t]; __syncthreads();
    for (int s = 128; s > 0; s >>= 1) { if (t < s) s_red[t] = fmaxf(s_red[t], s_red[t + s]); __syncthreads(); }
    float mu = s_red[0]; __syncthreads();
    s_red[t] = __expf(s_u[t] - mu); __syncthreads();
    for (int s = 128; s > 0; s >>= 1) { if (t < s) s_red[t] += s_red[t + s]; __syncthreads(); }
    float su = s_red[0]; __syncthreads();
    ws[WS_E + t] = __expf(s_u[t] - mu) / su;

    s_red[t] = s_v[t]; __syncthreads();
    for (int s = 128; s > 0; s >>= 1) { if (t < s) s_red[t] = fmaxf(s_red[t], s_red[t + s]); __syncthreads(); }
    float mv = s_red[0]; __syncthreads();
    s_red[t] = __expf(s_v[t] - mv); __syncthreads();
    for (int s = 128; s > 0; s >>= 1) { if (t < s) s_red[t] += s_red[t + s]; __syncthreads(); }
    float sv = s_red[0]; __syncthreads();
    ws[WS_F + t] = __expf(s_v[t] - mv) / sv;
}

// ---------------------------------------------------------------------------
// Kernel 2: materialize x = concat_n[obs_n | g | p_n] (114688 floats) and
// zero the fc1 accumulator. Block n builds row n with 448 threads.
// ---------------------------------------------------------------------------
__global__ __launch_bounds__(448) void k_build_x(const float* __restrict__ obs,
                                                 float* __restrict__ ws)
{
    const int n = blockIdx.x, t = threadIdx.x;
    float* x = ws + WS_X + (size_t)n * (DOBS + HSZ + NOBS);
    if (t < DOBS)            x[t] = obs[n * DOBS + t];
    else if (t < DOBS + HSZ) x[t] = ws[WS_G + (t - DOBS)];
    else                     x[t] = ws[WS_E + n] * ws[WS_F + (t - DOBS - HSZ)];
    if (n == 0) for (int i = t; i < F3H; i += 448) ws[WS_H1 + i] = 0.0f;
}

// ---------------------------------------------------------------------------
// Kernel 3: h1raw += W1 @ x via V_WMMA_F32_16X16X4_F32.
// B columns carry x[k] replicated -> every D column equals the matvec partial.
// grid = (32 row-blocks, 56 K-splits), 8 waves/WG, 256 K per wave (64 WMMAs).
// Per WMMA a wave streams 256B of W1 (16 sequential row-streams, prefetched).
// ---------------------------------------------------------------------------
__global__ __launch_bounds__(256) void k_fc1_wmma(const float* __restrict__ W1,
                                                  float* __restrict__ ws)
{
    __shared__ float red[8 * 16];
    const float* x  = ws + WS_X;
    float*       h1 = ws + WS_H1;

    const int t    = threadIdx.x;
    const int wave = t >> 5;
    const int lane = t & 31;
    const int rb   = blockIdx.x;                 // 16-row block: 0..31
    const int row  = rb * 16 + (lane & 15);      // A row this lane feeds
    const int hi   = lane >> 4;                  // 0: K={0,1}, 1: K={2,3}
    const int kw   = blockIdx.y * 8 + wave;      // 0..447
    const int kb   = kw * (F3IN / (KSPLIT * 8)); // 256-wide K chunk

    const float* pA = W1 + (size_t)row * F3IN + kb + 2 * hi;
    const float* pB = x + kb + 2 * hi;

    v8f c = {0.f, 0.f, 0.f, 0.f, 0.f, 0.f, 0.f, 0.f};
    for (int i = 0; i < 64; ++i) {
        v2f a = *(const v2f*)pA;                 // 16x4 fp32 A tile (ISA layout)
        v2f b = *(const v2f*)pB;                 // x[k] replicated across N cols
        __builtin_prefetch((const char*)pA + 1024, 0, 1);  // global_prefetch_b8
        c = __builtin_amdgcn_wmma_f32_16x16x4_f32(
                /*neg_a=*/false, a, /*neg_b=*/false, b,
                /*c_mod=*/(short)0, c, /*reuse_a=*/false, /*reuse_b=*/false);
        pA += 4; pB += 4;
    }

    // D columns are identical; N=0 lives in lanes 0 (M=0..7) and 16 (M=8..15)
    if (lane == 0)  for (int r = 0; r < 8; ++r) red[wave * 16 + r]     = c[r];
    if (lane == 16) for (int r = 0; r < 8; ++r) red[wave * 16 + 8 + r] = c[r];
    __syncthreads();
    if (t < 16) {
        float s = 0.0f;
        for (int w = 0; w < 8; ++w) s += red[w * 16 + t];
        atomicAdd(&h1[rb * 16 + t], s);
    }
}

// ---------------------------------------------------------------------------
// Kernel 4: h1 = relu(h1raw + b1); out = W2 @ h1 + b2   (40x512, tiny)
// ---------------------------------------------------------------------------
__global__ __launch_bounds__(512) void k_head(const float* __restrict__ b1,
                                              const float* __restrict__ W2,
                                              const float* __restrict__ b2,
                                              const float* __restrict__ ws,
                                              float* __restrict__ out)
{
    __shared__ float hr[F3H];
    const int t = threadIdx.x;
    float v = ws[WS_H1 + t] + b1[t];
    hr[t] = v > 0.0f ? v : 0.0f;
    __syncthreads();
    if (t < ADIM) {
        float acc = b2[t];
        for (int j = 0; j < F3H; ++j) acc += W2[t * F3H + j] * hr[j];
        out[t] = acc;
    }
}

// ---------------------------------------------------------------------------
extern "C" void kernel_launch(void* const* d_in, const int* in_sizes, int n_in,
                              void* d_out, int out_size, void* d_ws, size_t ws_size,
                              hipStream_t stream) {
    const float* obs  = (const float*)d_in[0];
    const int*   tok  = (const int*)  d_in[1];
    const float* embT = (const float*)d_in[2];
    const float* Wih  = (const float*)d_in[3];
    const float* Whh  = (const float*)d_in[4];
    const float* bih  = (const float*)d_in[5];
    const float* bhh  = (const float*)d_in[6];
    const float* f1W  = (const float*)d_in[7];
    // d_in[8] = f1_b: constant shift cancels in the full softmax -> unused
    const float* W1   = (const float*)d_in[9];
    const float* b1   = (const float*)d_in[10];
    const float* W2   = (const float*)d_in[11];
    const float* b2   = (const float*)d_in[12];
    float* ws  = (float*)d_ws;
    float* out = (float*)d_out;

    k_encoder <<<1, 256, 0, stream>>>(obs, tok, embT, Wih, Whh, bih, bhh, f1W, ws);
    k_build_x <<<NOBS, 448, 0, stream>>>(obs, ws);
    k_fc1_wmma<<<dim3(32, KSPLIT), 256, 0, stream>>>(W1, ws);
    k_head    <<<1, 512, 0, stream>>>(b1, W2, b2, ws, out);
}